// NodeGraphConvlutionalLayer_59279138619795
// MI455X (gfx1250) — compile-verified
//
#include <hip/hip_runtime.h>

// Problem constants (from the reference)
#define BB    8
#define NN    1024
#define EE    1536
#define FIN   64
#define FOUT  64
#define FEDGE 32
#define MAXDEG 256   // max incident edges per node we track (avg degree = 2E/N = 3)

typedef __attribute__((ext_vector_type(2))) float v2f;
typedef __attribute__((ext_vector_type(8))) float v8f;

// ---------------------------------------------------------------------------
// Kernel 1: ew[b,e] = sum_f edges[b,e,f] * wvec[f]
// ---------------------------------------------------------------------------
__global__ void k_edge_reduce(const float* __restrict__ edges,
                              const float* __restrict__ wvec,
                              float* __restrict__ ew) {
    int idx = blockIdx.x * blockDim.x + threadIdx.x;     // b*E + e
    if (idx >= BB * EE) return;
    const float* ep = edges + (size_t)idx * FEDGE;
    float acc = 0.f;
#pragma unroll
    for (int f = 0; f < FEDGE; ++f) acc += ep[f] * wvec[f];
    ew[idx] = acc;
}

// ---------------------------------------------------------------------------
// Kernel 2: endpoints of each edge from the incidence matrix.
// Thread e scans column e of inc[N,E]; consecutive threads hit consecutive
// addresses per row -> fully coalesced.
// ---------------------------------------------------------------------------
__global__ void k_endpoints(const float* __restrict__ inc,
                            int* __restrict__ s, int* __restrict__ d) {
    int e = blockIdx.x * blockDim.x + threadIdx.x;
    if (e >= EE) return;
    int first = -1, second = -1;
    for (int i = 0; i < NN; ++i) {
        if (inc[(size_t)i * EE + e] != 0.f) {
            if (first < 0) first = i;
            else { second = i; break; }
        }
    }
    s[e] = first;
    d[e] = second;
}

// ---------------------------------------------------------------------------
// Kernel 3: H = nodes @ W  ([B*N, 64] x [64, 64]) in full f32 precision via
// V_WMMA_F32_16X16X4_F32. One wave produces one 16x16 output tile; the K=64
// contraction is 16 chained WMMAs.
//
// Fragment layouts per CDNA5 ISA 7.12.2 (32-bit data, wave32):
//   A 16x4 (MxK): lanes 0-15 hold row M=lane, VGPR0/1 = K0/K1;
//                 lanes 16-31 hold same rows, VGPR0/1 = K2/K3.
//   B 4x16 (KxN): one row striped across lanes within a VGPR:
//                 VGPR0: lanes0-15 = K0 row, lanes16-31 = K2 row;
//                 VGPR1: lanes0-15 = K1 row, lanes16-31 = K3 row.
//   C/D 16x16:    VGPR v: lanes0-15 -> (M = v, N = lane),
//                 lanes16-31 -> (M = v+8, N = lane-16).
// ---------------------------------------------------------------------------
__global__ void k_hmat_wmma(const float* __restrict__ A,   // [8192, 64] = nodes
                            const float* __restrict__ W,   // [64, 64]
                            float* __restrict__ H) {       // [8192, 64]
    int wave  = (blockIdx.x * blockDim.x + threadIdx.x) >> 5;
    int lane  = threadIdx.x & 31;
    int mtile = wave >> 2;          // 512 M-tiles
    int ntile = wave & 3;           // 4  N-tiles
    int m0 = mtile * 16, n0 = ntile * 16;
    int hl   = lane & 15;
    int koff = (lane >> 4) ? 2 : 0;
    int rowA = m0 + hl;

    v8f c = {};
#pragma unroll
    for (int k = 0; k < 64; k += 4) {
        int kb = k + koff;
        v2f a, b;
        a.x = A[(size_t)rowA * 64 + kb];
        a.y = A[(size_t)rowA * 64 + kb + 1];
        b.x = W[(size_t)kb * 64 + n0 + hl];
        b.y = W[(size_t)(kb + 1) * 64 + n0 + hl];
        // 8 args: (neg_a, A, neg_b, B, c_mod, C, reuse_a, reuse_b)
        c = __builtin_amdgcn_wmma_f32_16x16x4_f32(
                false, a, false, b, (short)0, c, false, false);
    }

    int rbase = m0 + ((lane >> 4) ? 8 : 0);
#pragma unroll
    for (int v = 0; v < 8; ++v)
        H[(size_t)(rbase + v) * 64 + n0 + hl] = c[v];
}

// ---------------------------------------------------------------------------
// Kernel 4: sparse gather producing the final output. One block per node i,
// 64 threads = one output feature each. Edge list is built in a fixed order
// (single scan of row i of inc), and all reductions run in that fixed order,
// so results are bitwise deterministic across calls.
//
// out[b,i,f] = lap[i,i] * (sum_{e inc i} ew[b,e]) * H[b,i,f]
//            + sum_{e inc i} lap[i, j_e] * ew[b,e] * H[b, j_e, f]
// ---------------------------------------------------------------------------
__global__ void k_gather(const float* __restrict__ inc,
                         const float* __restrict__ lap,
                         const int* __restrict__ s,
                         const int* __restrict__ d,
                         const float* __restrict__ ew,
                         const float* __restrict__ H,
                         float* __restrict__ out) {
    __shared__ int   le[MAXDEG];
    __shared__ int   lj[MAXDEG];
    __shared__ float lw[MAXDEG];
    __shared__ int   sdeg;
    __shared__ float sdiag;

    int i = blockIdx.x;
    int f = threadIdx.x;            // 0..63

    if (threadIdx.x == 0) {
        int deg = 0;
        const float* row = inc + (size_t)i * EE;
        for (int e = 0; e < EE; ++e) {
            if (row[e] != 0.f && deg < MAXDEG) {
                int j = (s[e] == i) ? d[e] : s[e];
                le[deg] = e;
                lj[deg] = j;
                lw[deg] = lap[(size_t)i * NN + j];
                ++deg;
            }
        }
        sdeg = deg;
    }
    __syncthreads();

    int   deg = sdeg;
    float lii = lap[(size_t)i * NN + i];

    for (int b = 0; b < BB; ++b) {
        if (threadIdx.x == 0) {
            float ds = 0.f;
            for (int k = 0; k < deg; ++k) ds += ew[(size_t)b * EE + le[k]];
            sdiag = ds;
        }
        __syncthreads();

        float acc = lii * sdiag * H[((size_t)b * NN + i) * 64 + f];
        for (int k = 0; k < deg; ++k) {
            float coef = lw[k] * ew[(size_t)b * EE + le[k]];
            acc += coef * H[((size_t)b * NN + lj[k]) * 64 + f];
        }
        out[((size_t)b * NN + i) * 64 + f] = acc;
        __syncthreads();   // protect sdiag before next batch overwrites it
    }
}

// ---------------------------------------------------------------------------
// Launch. Inputs (setup_inputs order, all f32):
//   0 nodes [B,N,64]  1 edges [B,E,32]  2 W [64,64]
//   3 wvec [32]       4 inc [N,E]       5 lap [N,N]
// Workspace layout (needs ~2.06 MB):
//   ew: B*E floats @ 0      (49152 B)
//   s : E ints     @ 49152  ( 6144 B)
//   d : E ints     @ 55296  ( 6144 B)
//   H : B*N*64 f32 @ 61440  (2097152 B)
// ---------------------------------------------------------------------------
extern "C" void kernel_launch(void* const* d_in, const int* in_sizes, int n_in,
                              void* d_out, int out_size, void* d_ws, size_t ws_size,
                              hipStream_t stream) {
    const float* nodes = (const float*)d_in[0];
    const float* edges = (const float*)d_in[1];
    const float* W     = (const float*)d_in[2];
    const float* wvec  = (const float*)d_in[3];
    const float* inc   = (const float*)d_in[4];
    const float* lap   = (const float*)d_in[5];
    float* out = (float*)d_out;

    char* base = (char*)d_ws;
    float* ew = (float*)(base);
    int*   s  = (int*)(base + 49152);
    int*   d  = (int*)(base + 55296);
    float* H  = (float*)(base + 61440);

    // 1) edge feature reduction
    k_edge_reduce<<<(BB * EE + 255) / 256, 256, 0, stream>>>(edges, wvec, ew);

    // 2) edge endpoints from incidence matrix
    k_endpoints<<<(EE + 255) / 256, 256, 0, stream>>>(inc, s, d);

    // 3) H = nodes @ W  (2048 tiles, 8 waves per 256-thread block -> 256 blocks)
    k_hmat_wmma<<<256, 256, 0, stream>>>(nodes, W, H);

    // 4) sparse output gather, one block per node
    k_gather<<<NN, 64, 0, stream>>>(inc, lap, s, d, ew, H, out);
}